// PoincareEmbedding_13185549599244
// MI455X (gfx1250) — compile-verified
//
#include <hip/hip_runtime.h>

#define S_TOTAL 100000
#define DIMK 16
#define BATCH 2048
#define NSPLIT 10
#define SPLIT_LEN (S_TOTAL / NSPLIT)        // 10000
#define TILES_PER_SPLIT (SPLIT_LEN / 16)    // 625
#define NWAVES 8
#define BOUNDARY 0.99999f                   // 1 - 1e-5

typedef __attribute__((ext_vector_type(2))) float v2f;
typedef __attribute__((ext_vector_type(8))) float v8f;

__device__ __forceinline__ v8f wmma_f32_k4(v2f a, v2f b, v8f c) {
  // D = A(16x4 f32) * B(4x16 f32) + C(16x16 f32)
  return __builtin_amdgcn_wmma_f32_16x16x4_f32(false, a, false, b, (short)0, c,
                                               false, false);
}

// ---- one-time per-table precompute: swb[s] = { ||w_s||^2, 1/beta_s } ----
__global__ __launch_bounds__(256) void precompute_swb(
    const float* __restrict__ weight, v2f* __restrict__ swb) {
  int s = blockIdx.x * 256 + threadIdx.x;
  if (s < S_TOTAL) {
    const float* wr = weight + s * DIMK;
    float sq = 0.0f;
#pragma unroll
    for (int d = 0; d < DIMK; ++d) sq += wr[d] * wr[d];
    float beta = 1.0f - fminf(fmaxf(sq, 0.0f), BOUNDARY);
    v2f o;
    o.x = sq;
    o.y = 1.0f / beta;  // precise; executed once per table row
    swb[s] = o;
  }
}

// PASS 0: accumulate per-row sums of p = exp(-arccosh(x)) = 1/(x+sqrt(x^2-1))
// PASS 1: recompute p and write p * (1/rowsum) to out
template <int PASS>
__global__ __launch_bounds__(256) void poincare_pass(
    const int* __restrict__ idx, const float* __restrict__ weight,
    const v2f* __restrict__ swb, float* __restrict__ out,
    float* __restrict__ normbuf) {
  __shared__ float uT[16 * 18];        // u tile, padded stride
  __shared__ float sSq[16];            // ||u||^2 per row
  __shared__ float sTwoInvAl[16];      // 2 / alpha per row
  __shared__ float sInv[16];           // pass1: 1/rowsum broadcast
  __shared__ float wsums[NWAVES][16];  // pass0: per-wave row partials

  const int tid   = threadIdx.x;
  const int btile = blockIdx.x;        // 0..127  (16 batch rows each)
  const int split = blockIdx.y;        // 0..NSPLIT-1
  const int yoff  = split * SPLIT_LEN;

  // ---- stage u tile (16 rows x 16 dims) into LDS ----
  {
    int r = tid >> 4, c = tid & 15;
    int row = idx[btile * 16 + r];
    uT[r * 18 + c] = weight[row * DIMK + c];
  }
  if (PASS == 1 && tid < 16) sInv[tid] = normbuf[btile * 16 + tid];
  __syncthreads();
  if (tid < 16) {
    float s = 0.0f;
#pragma unroll
    for (int d = 0; d < DIMK; ++d) {
      float v = uT[tid * 18 + d];
      s += v * v;
    }
    sSq[tid] = s;
    float al = 1.0f - fminf(fmaxf(s, 0.0f), BOUNDARY);
    sTwoInvAl[tid] = 2.0f / al;  // once per row, precise
  }
  __syncthreads();

  // wave id as an SGPR so the main loop is scalar-controlled (EXEC stays
  // all-ones through the WMMA body, no divergent-loop exec accumulation)
  const int wave = __builtin_amdgcn_readfirstlane(tid >> 5);
  const int lane = tid & 31;
  const int n    = lane & 15;   // WMMA column / A-row index
  const int kh   = lane >> 4;   // K-half select

  // ---- A fragments (u), ISA fp32 16x4 layout: lane holds K = 4k+2kh, +1 ----
  v2f a[4];
#pragma unroll
  for (int k = 0; k < 4; ++k) {
    a[k].x = uT[n * 18 + 4 * k + 2 * kh];
    a[k].y = uT[n * 18 + 4 * k + 2 * kh + 1];
  }

  // per-lane row constants for C/D rows m = r + 8*kh
  float squ[8], tia[8], invRow[8];
#pragma unroll
  for (int r = 0; r < 8; ++r) {
    squ[r] = sSq[r + 8 * kh];
    tia[r] = sTwoInvAl[r + 8 * kh];
    if (PASS == 1) invRow[r] = sInv[r + 8 * kh];
  }

  float acc[8];
#pragma unroll
  for (int r = 0; r < 8; ++r) acc[r] = 0.0f;

  // ---- main loop: each wave owns a strided set of 16-wide S tiles ----
  for (int t = wave; t < TILES_PER_SPLIT; t += NWAVES) {
    const int sbase = yoff + t * 16;
    const float* wr = weight + (sbase + n) * DIMK + 2 * kh;

    // branchless clamped prefetch of this wave's next tile (uniform cselect;
    // never touches past the table)
    {
      int pft = t + NWAVES;
      int pfs = (pft < TILES_PER_SPLIT) ? (yoff + pft * 16) : yoff;
      __builtin_prefetch(weight + (pfs + n) * DIMK, 0, 1);
    }

    // B fragments (w^T), fp32 4x16 layout: lane holds K = 4k+2kh, +1 of col n
    v2f b0 = *(const v2f*)(wr + 0);
    v2f b1 = *(const v2f*)(wr + 4);
    v2f b2 = *(const v2f*)(wr + 8);
    v2f b3 = *(const v2f*)(wr + 12);

    // precomputed { ||w_s||^2, 1/beta_s } for this lane's column
    v2f sb = swb[sbase + n];
    const float sqw = sb.x;
    const float rb  = sb.y;

    v8f c = {};
    c = wmma_f32_k4(a[0], b0, c);
    c = wmma_f32_k4(a[1], b1, c);
    c = wmma_f32_k4(a[2], b2, c);
    c = wmma_f32_k4(a[3], b3, c);

#pragma unroll
    for (int r = 0; r < 8; ++r) {
      float cr = c[r];
      float sqd = fmaxf(squ[r] + sqw - 2.0f * cr, 0.0f);
      float x = fmaf(sqd, tia[r] * rb, 1.0f);  // 2*sqd/(alpha*beta) + 1
      float y = x + __builtin_amdgcn_sqrtf(fmaxf(fmaf(x, x, -1.0f), 0.0f));
      float p = __builtin_amdgcn_rcpf(y);  // exp(-arccosh(x)), GAMMA==1
      if (PASS == 0) {
        acc[r] += p;
      } else {
        size_t o = (size_t)(btile * 16 + r + 8 * kh) * S_TOTAL + (sbase + n);
        __builtin_nontemporal_store(p * invRow[r], &out[o]);
      }
    }
  }

  if (PASS == 0) {
    // reduce over the 16 lanes of each half-wave (columns of this wave)
#pragma unroll
    for (int r = 0; r < 8; ++r) {
#pragma unroll
      for (int off = 1; off < 16; off <<= 1)
        acc[r] += __shfl_xor(acc[r], off, 32);
    }
    if (n == 0) {
#pragma unroll
      for (int r = 0; r < 8; ++r) wsums[wave][r + 8 * kh] = acc[r];
    }
    __syncthreads();
    if (tid < 16) {
      float s = 0.0f;  // fixed-order (deterministic) cross-wave reduction
#pragma unroll
      for (int w = 0; w < NWAVES; ++w) s += wsums[w][tid];
      normbuf[(btile * 16 + tid) * NSPLIT + split] = s;
    }
  }
}

__global__ __launch_bounds__(256) void row_norm_kernel(
    const float* __restrict__ partial, float* __restrict__ inv) {
  int row = blockIdx.x * blockDim.x + threadIdx.x;
  if (row < BATCH) {
    float s = 0.0f;
#pragma unroll
    for (int k = 0; k < NSPLIT; ++k) s += partial[row * NSPLIT + k];
    inv[row] = 1.0f / s;
  }
}

extern "C" void kernel_launch(void* const* d_in, const int* in_sizes, int n_in,
                              void* d_out, int out_size, void* d_ws,
                              size_t ws_size, hipStream_t stream) {
  const int* idx      = (const int*)d_in[0];
  const float* weight = (const float*)d_in[1];
  float* out          = (float*)d_out;

  // workspace layout: partial[BATCH*NSPLIT] | inv[BATCH] | swb[S_TOTAL] (v2f)
  float* partial = (float*)d_ws;
  float* inv     = partial + BATCH * NSPLIT;
  v2f* swb       = (v2f*)(inv + BATCH);  // 8-byte aligned (offset 90112)

  precompute_swb<<<(S_TOTAL + 255) / 256, 256, 0, stream>>>(weight, swb);

  dim3 grid(BATCH / 16, NSPLIT);
  poincare_pass<0><<<grid, 256, 0, stream>>>(idx, weight, swb, out, partial);
  row_norm_kernel<<<(BATCH + 255) / 256, 256, 0, stream>>>(partial, inv);
  poincare_pass<1><<<grid, 256, 0, stream>>>(idx, weight, swb, out, inv);
}